// PhysicsNet_80401787781597
// MI455X (gfx1250) — compile-verified
//
#include <hip/hip_runtime.h>

// One QP instance per lane. 20 IPM iterations, Schur-complement 6x6 Cholesky
// per iteration, all state in VGPRs. Native v_rcp_f32/v_rsq_f32 (TRANS ops,
// co-issue with VALU) replace IEEE div/sqrt sequences in the latency-bound
// loop. Epilogue computes the batched Q@Z for the cost term with
// V_WMMA_F32_16X16X4_F32 (two K-chunks per 16-instance tile), staged through
// LDS to match the WMMA B/D lane layouts.

typedef float v2f __attribute__((ext_vector_type(2)));
typedef float v8f __attribute__((ext_vector_type(8)));

__device__ __forceinline__ float fast_rcp(float x)  { return __builtin_amdgcn_rcpf(x); }
__device__ __forceinline__ float fast_rsqrt(float x){ return __builtin_amdgcn_rsqf(x); }

// Qmod padded to 16 rows (WMMA M dim), 6 K-columns.
__constant__ float c_Qpad[16][6] = {
    { 4.001f, -4.0f,   0.0f,   0.0f,   0.0f,   0.0f },
    { -4.0f,  4.001f,  0.0f,   0.0f,   0.0f,   0.0f },
    { 0.0f,   0.0f,    0.001f, 0.0f,   0.0f,   0.0f },
    { 0.0f,   0.0f,    0.0f,   0.001f, 0.0f,   0.0f },
    { 0.0f,   0.0f,    0.0f,   0.0f,   0.001f, 0.0f },
    { 0.0f,   0.0f,    0.0f,   0.0f,   0.0f,   0.001f },
    {0,0,0,0,0,0},{0,0,0,0,0,0},{0,0,0,0,0,0},{0,0,0,0,0,0},
    {0,0,0,0,0,0},{0,0,0,0,0,0},{0,0,0,0,0,0},{0,0,0,0,0,0},
    {0,0,0,0,0,0},{0,0,0,0,0,0}
};

__global__ __launch_bounds__(256) void qp_ipm_kernel(
    const float* __restrict__ vk, const float* __restrict__ mu_p,
    float* __restrict__ out, int Btot)
{
    __shared__ float zs[6][256];   // z components, instance-major across block
    __shared__ float qz[6][256];   // Q@z results

    const int tid = threadIdx.x;
    const int gid = blockIdx.x * 256 + tid;

    const float v  = (gid < Btot) ? vk[gid] : 0.0f;
    const float mu = mu_p[0];

    // beta = next_vel - v - u = 2 - v - 2 = -v ; vu = v + u = v + 2
    const float beta = -v;
    const float vu   = v + 2.0f;

    // p = b + fpad + slack_penalty
    float p[6];
    p[0] =  3.0f * v + 2.0f;     // -2*beta + vu = 2v + v + 2
    p[1] = -3.0f * v - 2.0f;
    p[2] =  mu;
    p[3] = 1.0f; p[4] = 1.0f; p[5] = 1.0f;

    float h[9] = {0,0,0,0,0,0, vu, -vu, mu};

    // Constant structure: Qmod (sparse), Rg = [-G | I3], R = [-I6 ; Rg]
    const float Qm[6][6] = {
        { 4.001f, -4.0f,   0.0f,   0.0f,   0.0f,   0.0f },
        { -4.0f,  4.001f,  0.0f,   0.0f,   0.0f,   0.0f },
        { 0.0f,   0.0f,    0.001f, 0.0f,   0.0f,   0.0f },
        { 0.0f,   0.0f,    0.0f,   0.001f, 0.0f,   0.0f },
        { 0.0f,   0.0f,    0.0f,   0.0f,   0.001f, 0.0f },
        { 0.0f,   0.0f,    0.0f,   0.0f,   0.0f,   0.001f }
    };
    const float Rg[3][6] = {  // rows 6..8 of R: [-G | I3]
        { -1.0f,  1.0f, -1.0f, 1.0f, 0.0f, 0.0f },
        {  1.0f, -1.0f, -1.0f, 0.0f, 1.0f, 0.0f },
        {  1.0f,  1.0f,  0.0f, 0.0f, 0.0f, 1.0f }
    };

    float z[6] = {0,0,0,0,0,0};
    float s[9], lam[9];
#pragma unroll
    for (int j = 0; j < 9; ++j) { s[j] = 1.0f; lam[j] = 1.0f; }

#pragma unroll 1
    for (int it = 0; it < 20; ++it) {
        float invlam[9], w[9];
#pragma unroll
        for (int j = 0; j < 9; ++j) {
            invlam[j] = fast_rcp(lam[j]);
            w[j]      = lam[j] * fast_rcp(s[j]);
        }

        // Q @ z (sparse Q)
        float qzv[6];
        qzv[0] =  4.001f * z[0] - 4.0f * z[1];
        qzv[1] = -4.0f * z[0] + 4.001f * z[1];
#pragma unroll
        for (int j = 2; j < 6; ++j) qzv[j] = 0.001f * z[j];

        // r_d = Qz + p + R^T lam   (R^T lam: -lam[a] + sum_k Rg[k][a]*lam[6+k])
        float rd[6];
#pragma unroll
        for (int a = 0; a < 6; ++a) {
            float t = qzv[a] + p[a] - lam[a];
#pragma unroll
            for (int k = 0; k < 3; ++k) t += Rg[k][a] * lam[6 + k];
            rd[a] = t;
        }

        // r_p = R z + s - h
        float rp[9];
#pragma unroll
        for (int j = 0; j < 6; ++j) rp[j] = s[j] - z[j];          // h[0..5]=0
#pragma unroll
        for (int k = 0; k < 3; ++k) {
            float t = s[6 + k] - h[6 + k];
#pragma unroll
            for (int a = 0; a < 6; ++a) t += Rg[k][a] * z[a];
            rp[6 + k] = t;
        }

        // mu_bar, complementarity residual, condensed rhs
        float dotsl = 0.0f;
#pragma unroll
        for (int j = 0; j < 9; ++j) dotsl += s[j] * lam[j];
        const float mubar = 0.1f * dotsl * (1.0f / 9.0f);

        float comp[9], rhs2[9];
#pragma unroll
        for (int j = 0; j < 9; ++j) {
            comp[j] = mubar - s[j] * lam[j];
            rhs2[j] = -rp[j] - comp[j] * invlam[j];
        }

        // Schur matrix  K = Q + diag(w[0..5]) + Rg^T diag(w[6..8]) Rg  (SPD)
        float K[6][6];
#pragma unroll
        for (int a = 0; a < 6; ++a) {
#pragma unroll
            for (int b = 0; b <= a; ++b) {
                float acc = Qm[a][b];
                if (a == b) acc += w[a];
#pragma unroll
                for (int k = 0; k < 3; ++k) acc += w[6 + k] * Rg[k][a] * Rg[k][b];
                K[a][b] = acc;
            }
        }

        // g = -r_d + R^T (w .* rhs2)
        float g[6];
#pragma unroll
        for (int a = 0; a < 6; ++a) {
            float t = -rd[a] - w[a] * rhs2[a];
#pragma unroll
            for (int k = 0; k < 3; ++k) t += Rg[k][a] * w[6 + k] * rhs2[6 + k];
            g[a] = t;
        }

        // In-place 6x6 Cholesky (reciprocal diag via v_rsq), two triangular solves
#pragma unroll
        for (int j = 0; j < 6; ++j) {
            float d = K[j][j];
#pragma unroll
            for (int k = 0; k < j; ++k) d -= K[j][k] * K[j][k];
            const float rinv = fast_rsqrt(d);
#pragma unroll
            for (int i = j + 1; i < 6; ++i) {
                float t = K[i][j];
#pragma unroll
                for (int k = 0; k < j; ++k) t -= K[i][k] * K[j][k];
                K[i][j] = t * rinv;
            }
            K[j][j] = rinv;
        }
        float y[6], dz[6];
#pragma unroll
        for (int j = 0; j < 6; ++j) {
            float t = g[j];
#pragma unroll
            for (int k = 0; k < j; ++k) t -= K[j][k] * y[k];
            y[j] = t * K[j][j];
        }
#pragma unroll
        for (int j = 5; j >= 0; --j) {
            float t = y[j];
#pragma unroll
            for (int k = j + 1; k < 6; ++k) t -= K[k][j] * dz[k];
            dz[j] = t * K[j][j];
        }

        // dlam = w .* (R dz - rhs2) ; ds = (comp - s.*dlam) ./ lam
        float Rdz[9];
#pragma unroll
        for (int j = 0; j < 6; ++j) Rdz[j] = -dz[j];
#pragma unroll
        for (int k = 0; k < 3; ++k) {
            float t = 0.0f;
#pragma unroll
            for (int a = 0; a < 6; ++a) t += Rg[k][a] * dz[a];
            Rdz[6 + k] = t;
        }
        float dlam[9], dsv[9];
#pragma unroll
        for (int j = 0; j < 9; ++j) {
            dlam[j] = w[j] * (Rdz[j] - rhs2[j]);
            dsv[j]  = (comp[j] - s[j] * dlam[j]) * invlam[j];
        }

        // alpha = 0.99 * min(maxstep(s,ds), maxstep(lam,dlam))
        float a1 = 1.0f, a2 = 1.0f;
#pragma unroll
        for (int j = 0; j < 9; ++j) {
            const float rs = -s[j]   * fast_rcp(dsv[j]);
            const float rl = -lam[j] * fast_rcp(dlam[j]);
            a1 = fminf(a1, (dsv[j]  < 0.0f) ? rs : 1.0e9f);
            a2 = fminf(a2, (dlam[j] < 0.0f) ? rl : 1.0e9f);
        }
        const float alpha = 0.99f * fminf(a1, a2);

#pragma unroll
        for (int j = 0; j < 6; ++j) z[j] += alpha * dz[j];
#pragma unroll
        for (int j = 0; j < 9; ++j) { s[j] += alpha * dsv[j]; lam[j] += alpha * dlam[j]; }
    }

    // ---- Epilogue: batched Q@Z via V_WMMA_F32_16X16X4_F32 -------------------
    // Stage z into LDS as a 6 x 256 tile, then each wave does two 16-instance
    // tiles: D(16x16) = Qpad(16x4-chunk) x Z(4x16-chunk), K split 4 + 2.
#pragma unroll
    for (int m = 0; m < 6; ++m) zs[m][tid] = z[m];
    __syncthreads();

    const int lane     = tid & 31;
    const int wavebase = tid & ~31;          // first column of this wave's tile
    const int mrow     = lane & 15;          // M index for A / N index for B
    const bool lohalf  = (lane < 16);
    const int kb       = lohalf ? 0 : 2;

    // Unconditional loads + v_cndmask masking (avoid exec-save branches).
    v2f A1, A2;
    A1.x = c_Qpad[mrow][kb];
    A1.y = c_Qpad[mrow][kb + 1];
    {
        const float q4 = c_Qpad[mrow][4];
        const float q5 = c_Qpad[mrow][5];
        A2.x = lohalf ? q4 : 0.0f;           // K-slots 2,3 of chunk 2 are zero
        A2.y = lohalf ? q5 : 0.0f;
    }

#pragma unroll
    for (int t = 0; t < 2; ++t) {
        const int col = wavebase + t * 16 + mrow;
        v2f B1, B2;
        B1.x = zs[kb][col];
        B1.y = zs[kb + 1][col];
        const float z4 = zs[4][col];
        const float z5 = zs[5][col];
        B2.x = lohalf ? z4 : 0.0f;
        B2.y = lohalf ? z5 : 0.0f;

        v8f c = {};
        // D = A1*B1 + A2*B2  (accumulate chain; exact f32 MACs)
        c = __builtin_amdgcn_wmma_f32_16x16x4_f32(false, A2, false, B2,
                                                  (short)0, c, false, false);
        c = __builtin_amdgcn_wmma_f32_16x16x4_f32(false, A1, false, B1,
                                                  (short)0, c, false, false);
        if (lohalf) {
#pragma unroll
            for (int m = 0; m < 6; ++m) qz[m][col] = c[m];  // rows 0..5 live in lanes 0-15
        }
    }
    __syncthreads();

    float zq = 0.0f, pz = 0.0f;
#pragma unroll
    for (int m = 0; m < 6; ++m) {
        zq += z[m] * qz[m][tid];
        pz += p[m] * z[m];
    }
    if (gid < Btot) out[gid] = 0.5f * zq + pz + beta * beta;
}

extern "C" void kernel_launch(void* const* d_in, const int* in_sizes, int n_in,
                              void* d_out, int out_size, void* d_ws, size_t ws_size,
                              hipStream_t stream) {
    (void)n_in; (void)out_size; (void)d_ws; (void)ws_size;
    const float* vk = (const float*)d_in[0];
    const float* mu = (const float*)d_in[1];
    float* out = (float*)d_out;
    const int Btot = in_sizes[0];
    const int grid = (Btot + 255) / 256;
    qp_ipm_kernel<<<grid, 256, 0, stream>>>(vk, mu, out, Btot);
}